// GFlowNetActor_62491774157490
// MI455X (gfx1250) — compile-verified
//
#include <hip/hip_runtime.h>
#include <float.h>
#include <math.h>
#include <stdint.h>

// ---------------------------------------------------------------------------
// GFlowNet actor log-prob: masked segment-logsumexp (E=8.4M edges, G=4096).
// Memory-bound: ~260MB of traffic -> ~11us at 23.3 TB/s. Strategy:
//  - TDM (tensor_load_to_lds) double-buffered staging of the edge streams,
//    synchronized with s_wait_tensorcnt (CDNA5 async-tensor path).
//  - Sorted segment ids -> per-block LDS window accumulators (2048 segs)
//    so global atomics are ~G-scale, not E-scale.
// ---------------------------------------------------------------------------

typedef __attribute__((ext_vector_type(4))) unsigned int v4u_t;
typedef __attribute__((ext_vector_type(4))) int          v4i_t;
typedef __attribute__((ext_vector_type(8))) int          v8i_t;

#define WAVE 32
#define BLOCK_THREADS 256
#define WAVES_PB 8
#define TILE_W 256                 // elements staged per wave per iteration
#define TILE_B (TILE_W * WAVES_PB) // 2048 per block iteration
#define BLOCK_EDGES 16384          // contiguous edges per block
#define WIN 2048                   // LDS segment-accumulator window

#define NEG_INF    (-3.4028234663852886e38f)  // float32 finfo.min
#define F32_EPS    (1.1920928955078125e-07f)  // float32 eps
#define ORD_NEGMAX 0x00800000u                // f2o(-FLT_MAX)

// Order-preserving float<->uint mapping so we can use integer atomicMax.
__device__ __forceinline__ unsigned f2o(float f) {
  unsigned u = __float_as_uint(f);
  return (u & 0x80000000u) ? ~u : (u | 0x80000000u);
}
__device__ __forceinline__ float o2f(unsigned o) {
  unsigned u = (o & 0x80000000u) ? (o & 0x7fffffffu) : ~o;
  return __uint_as_float(u);
}

// Byte offset of a __shared__ pointer within LDS (flat addr low 32 bits).
__device__ __forceinline__ unsigned lds_off(const void* p) {
  return (unsigned)(uintptr_t)p;
}

// Issue a 1-D TDM copy: nelem 32-bit elements from global -> LDS.
// Descriptor layout per CDNA5 ISA ch.8 (D# groups 0/1); groups 2/3 zero
// (2-D tensor with dim1 = 1). Tracked on TENSORcnt.
__device__ __forceinline__ void tdm_load_1d_b32(unsigned lds_byte_off,
                                                const void* gptr,
                                                unsigned nelem) {
  unsigned long long ga = (unsigned long long)(uintptr_t)gptr;
  v4u_t g0;
  g0.x = 1u;                                    // count=1, user descriptor
  g0.y = lds_byte_off;                          // lds_addr
  g0.z = (unsigned)ga;                          // global_addr[31:0]
  g0.w = ((unsigned)(ga >> 32) & 0x01FFFFFFu)   // global_addr[56:32]
         | 0x80000000u;                         // type = 2 ("image")
  v8i_t g1;
  g1[0] = (int)(2u << 16);                      // wg_mask=0, data_size=4B
  g1[1] = (int)((nelem & 0xFFFFu) << 16);       // tensor_dim0[15:0]
  g1[2] = (int)(((nelem >> 16) & 0xFFFFu)       // tensor_dim0[31:16]
                | (1u << 16));                  // tensor_dim1 = 1
  g1[3] = (int)((nelem & 0xFFFFu) << 16);       // tile_dim0 = nelem (<=64K)
  g1[4] = 1;                                    // tile_dim1 = 1, tile_dim2 = 0
  g1[5] = (int)nelem;                           // tensor_dim0_stride
  g1[6] = 0;
  g1[7] = 0;
  v4i_t z4 = {0, 0, 0, 0};
#if defined(__clang_major__) && (__clang_major__ >= 23)
  v8i_t z8 = {0, 0, 0, 0, 0, 0, 0, 0};
  __builtin_amdgcn_tensor_load_to_lds(g0, g1, z4, z4, z8, 0);
#else
  __builtin_amdgcn_tensor_load_to_lds(g0, g1, z4, z4, 0);
#endif
}

// ---------------------------------------------------------------------------
__global__ void init_ws_kernel(unsigned* __restrict__ g_max,
                               float* __restrict__ g_sum, int G) {
  int g = blockIdx.x * blockDim.x + threadIdx.x;
  if (g < G) {
    g_max[g] = ORD_NEGMAX;  // == f2o(float32 finfo.min)
    g_sum[g] = 0.0f;
  }
}

// ---------------------------------------------------------------------------
// Pass 1: per-segment max of masked logits.
__global__ __launch_bounds__(BLOCK_THREADS)
void pass1_segmax(const float* __restrict__ logits,
                  const int* __restrict__ batch,
                  const unsigned char* __restrict__ valid,
                  unsigned* __restrict__ g_max, int E, int G) {
  __shared__ float    s_log[WAVES_PB][2][TILE_W];
  __shared__ int      s_bat[WAVES_PB][2][TILE_W];
  __shared__ unsigned s_win[WIN];

  const int e0 = blockIdx.x * BLOCK_EDGES;
  const int e1 = min(e0 + BLOCK_EDGES, E);
  const int w = threadIdx.x >> 5;
  const int lane = threadIdx.x & (WAVE - 1);
  const int segLo = batch[e0];  // sorted -> minimum segment in this region

  for (int j = threadIdx.x; j < WIN; j += BLOCK_THREADS) s_win[j] = ORD_NEGMAX;
  __syncthreads();

  const int nIter = (e1 - e0 + TILE_B - 1) / TILE_B;

  // Prologue: stage iteration 0 (per-wave private LDS buffers -> no barrier).
  {
    int st = e0 + w * TILE_W;
    int n = min(TILE_W, e1 - st);
    if (n > 0) {
      tdm_load_1d_b32(lds_off(&s_log[w][0][0]), logits + st, (unsigned)n);
      tdm_load_1d_b32(lds_off(&s_bat[w][0][0]), batch + st, (unsigned)n);
    }
  }
  for (int it = 0; it < nIter; ++it) {
    const int buf = it & 1;
    bool issuedNext = false;
    if (it + 1 < nIter) {
      int st = e0 + (it + 1) * TILE_B + w * TILE_W;
      int n = min(TILE_W, e1 - st);
      if (n > 0) {
        tdm_load_1d_b32(lds_off(&s_log[w][buf ^ 1][0]), logits + st, (unsigned)n);
        tdm_load_1d_b32(lds_off(&s_bat[w][buf ^ 1][0]), batch + st, (unsigned)n);
        issuedNext = true;
      }
    }
    if (issuedNext) __builtin_amdgcn_s_wait_tensorcnt(2);  // current tile done
    else            __builtin_amdgcn_s_wait_tensorcnt(0);

    int st = e0 + it * TILE_B + w * TILE_W;
    int n = min(TILE_W, e1 - st);
    for (int k = lane; k < n; k += WAVE) {
      float x = s_log[w][buf][k];
      int s = s_bat[w][buf][k];
      float m = valid[st + k] ? x : NEG_INF;  // TEMP == 1.0
      unsigned o = f2o(m);
      unsigned rel = (unsigned)(s - segLo);
      if (rel < WIN) atomicMax(&s_win[rel], o);
      else if ((unsigned)s < (unsigned)G) atomicMax(&g_max[s], o);
    }
  }
  __syncthreads();
  for (int j = threadIdx.x; j < WIN; j += BLOCK_THREADS) {
    unsigned o = s_win[j];
    int s = segLo + j;
    if (o != ORD_NEGMAX && s < G) atomicMax(&g_max[s], o);
  }
}

// ---------------------------------------------------------------------------
// Pass 2: per-segment sum of exp(masked - segmax).
__global__ __launch_bounds__(BLOCK_THREADS)
void pass2_sumexp(const float* __restrict__ logits,
                  const int* __restrict__ batch,
                  const unsigned char* __restrict__ valid,
                  const unsigned* __restrict__ g_max,
                  float* __restrict__ g_sum, int E, int G) {
  __shared__ float s_log[WAVES_PB][2][TILE_W];
  __shared__ int   s_bat[WAVES_PB][2][TILE_W];
  __shared__ float s_sum[WIN];
  __shared__ float s_maxw[WIN];

  const int e0 = blockIdx.x * BLOCK_EDGES;
  const int e1 = min(e0 + BLOCK_EDGES, E);
  const int w = threadIdx.x >> 5;
  const int lane = threadIdx.x & (WAVE - 1);
  const int segLo = batch[e0];

  for (int j = threadIdx.x; j < WIN; j += BLOCK_THREADS) {
    s_sum[j] = 0.0f;
    int s = segLo + j;
    s_maxw[j] = (s < G) ? o2f(g_max[s]) : NEG_INF;
  }
  __syncthreads();

  const int nIter = (e1 - e0 + TILE_B - 1) / TILE_B;
  {
    int st = e0 + w * TILE_W;
    int n = min(TILE_W, e1 - st);
    if (n > 0) {
      tdm_load_1d_b32(lds_off(&s_log[w][0][0]), logits + st, (unsigned)n);
      tdm_load_1d_b32(lds_off(&s_bat[w][0][0]), batch + st, (unsigned)n);
    }
  }
  for (int it = 0; it < nIter; ++it) {
    const int buf = it & 1;
    bool issuedNext = false;
    if (it + 1 < nIter) {
      int st = e0 + (it + 1) * TILE_B + w * TILE_W;
      int n = min(TILE_W, e1 - st);
      if (n > 0) {
        tdm_load_1d_b32(lds_off(&s_log[w][buf ^ 1][0]), logits + st, (unsigned)n);
        tdm_load_1d_b32(lds_off(&s_bat[w][buf ^ 1][0]), batch + st, (unsigned)n);
        issuedNext = true;
      }
    }
    if (issuedNext) __builtin_amdgcn_s_wait_tensorcnt(2);
    else            __builtin_amdgcn_s_wait_tensorcnt(0);

    int st = e0 + it * TILE_B + w * TILE_W;
    int n = min(TILE_W, e1 - st);
    for (int k = lane; k < n; k += WAVE) {
      float x = s_log[w][buf][k];
      int s = s_bat[w][buf][k];
      float m = valid[st + k] ? x : NEG_INF;
      unsigned rel = (unsigned)(s - segLo);
      float mseg;
      if (rel < WIN) mseg = s_maxw[rel];
      else mseg = ((unsigned)s < (unsigned)G) ? o2f(g_max[s]) : 0.0f;
      // m == mseg == NEG_INF -> exp(0) = 1, matching the reference exactly.
      float sh = __expf(m - mseg);
      if (rel < WIN) atomicAdd(&s_sum[rel], sh);
      else if ((unsigned)s < (unsigned)G) atomicAdd(&g_sum[s], sh);
    }
  }
  __syncthreads();
  for (int j = threadIdx.x; j < WIN; j += BLOCK_THREADS) {
    float v = s_sum[j];
    int s = segLo + j;
    if (v != 0.0f && s < G) atomicAdd(&g_sum[s], v);
  }
}

// ---------------------------------------------------------------------------
// Finalize per-graph quantities.
__global__ void finalize_kernel(const unsigned* __restrict__ g_max,
                                const float* __restrict__ g_sum,
                                const float* __restrict__ stop_logits,
                                float* __restrict__ out_stop,
                                float* __restrict__ out_denom,
                                float* __restrict__ out_has, int G) {
  int g = blockIdx.x * blockDim.x + threadIdx.x;
  if (g >= G) return;
  float m = o2f(g_max[g]);
  float lse = logf(fmaxf(g_sum[g], F32_EPS)) + m;  // log(max(sum,eps)) + max
  float st = stop_logits[g];                        // TEMP == 1.0
  float a = fmaxf(lse, st);
  float b = fminf(lse, st);
  float ld = a + log1pf(__expf(b - a));             // logaddexp
  out_denom[g] = ld;
  out_stop[g] = st - ld;
  out_has[g] = (m > NEG_INF) ? 1.0f : 0.0f;
}

// ---------------------------------------------------------------------------
// Pass 3: per-edge log-prob, vectorized 128-bit streaming.
__global__ __launch_bounds__(256)
void pass3_edges(const float* __restrict__ logits,
                 const int* __restrict__ batch,
                 const unsigned char* __restrict__ valid,
                 const float* __restrict__ denom,
                 float* __restrict__ out_edge, int E) {
  int n4 = E >> 2;
  int stride = gridDim.x * blockDim.x;
  for (int i = blockIdx.x * blockDim.x + threadIdx.x; i < n4; i += stride) {
    float4 x = ((const float4*)logits)[i];
    int4 s = ((const int4*)batch)[i];
    unsigned vb = ((const unsigned*)valid)[i];
    float4 r;
    r.x = (((vb      ) & 0xFFu) ? x.x : NEG_INF) - denom[s.x];
    r.y = (((vb >>  8) & 0xFFu) ? x.y : NEG_INF) - denom[s.y];
    r.z = (((vb >> 16) & 0xFFu) ? x.z : NEG_INF) - denom[s.z];
    r.w = (((vb >> 24) & 0xFFu) ? x.w : NEG_INF) - denom[s.w];
    ((float4*)out_edge)[i] = r;
  }
  if (blockIdx.x == 0 && threadIdx.x == 0) {  // scalar tail (E % 4)
    for (int i = n4 << 2; i < E; ++i) {
      float m = valid[i] ? logits[i] : NEG_INF;
      out_edge[i] = m - denom[batch[i]];
    }
  }
}

// ---------------------------------------------------------------------------
extern "C" void kernel_launch(void* const* d_in, const int* in_sizes, int n_in,
                              void* d_out, int out_size, void* d_ws, size_t ws_size,
                              hipStream_t stream) {
  const float*         logits = (const float*)d_in[0];
  const float*         stop   = (const float*)d_in[1];
  const int*           batch  = (const int*)d_in[2];
  const unsigned char* valid  = (const unsigned char*)d_in[3];
  const int E = in_sizes[0];
  const int G = in_sizes[1];

  float* out_edge  = (float*)d_out;       // [E]
  float* out_stop  = out_edge + E;        // [G]
  float* out_denom = out_stop + G;        // [G]
  float* out_has   = out_denom + G;       // [G]

  unsigned* g_max = (unsigned*)d_ws;      // [G] ordered-float max
  float*    g_sum = (float*)(g_max + G);  // [G]

  const int gG = (G + 255) / 256;
  init_ws_kernel<<<gG, 256, 0, stream>>>(g_max, g_sum, G);

  const int nBlocks = (E + BLOCK_EDGES - 1) / BLOCK_EDGES;
  pass1_segmax<<<nBlocks, BLOCK_THREADS, 0, stream>>>(logits, batch, valid,
                                                      g_max, E, G);
  pass2_sumexp<<<nBlocks, BLOCK_THREADS, 0, stream>>>(logits, batch, valid,
                                                      g_max, g_sum, E, G);
  finalize_kernel<<<gG, 256, 0, stream>>>(g_max, g_sum, stop,
                                          out_stop, out_denom, out_has, G);
  int n4 = E >> 2;
  int g3 = (n4 + 255) / 256;
  if (g3 < 1) g3 = 1;
  if (g3 > 8192) g3 = 8192;  // grid-stride covers the rest
  pass3_edges<<<g3, 256, 0, stream>>>(logits, batch, valid, out_denom,
                                      out_edge, E);
}